// iidBindingKroneckerJoint_68839735820334
// MI455X (gfx1250) — compile-verified
//
#include <hip/hip_runtime.h>

typedef float v2f __attribute__((ext_vector_type(2)));
typedef float v8f __attribute__((ext_vector_type(8)));

#define KMAX 16
#define EPSF 1e-8f
#define TPB 256

struct Tabs {
  float a[KMAX + 1];     // sigmoid(alpha)
  float lf[KMAX + 1];    // log-factorials
  float pw00[KMAX + 1], pw01[KMAX + 1], pw11[KMAX + 1]; // s10 == s01
  float nf, inv_nf;
  int k;
};

__device__ inline void build_tabs(const float* alpha, float se00, float se01, float se11,
                                  float ssum_orig, int n_round, int k, Tabs& T) {
  float s_sum = se00 + 2.0f * se01 + se11;
  float ratio = ssum_orig / s_sum;
  float s00 = fminf(fmaxf(se00 * ratio, 1.0e-4f), 1.0f - 1.0e-4f);
  float s01 = fminf(fmaxf(se01 * ratio, 1.0e-4f), 1.0f - 1.0e-4f);
  float s11 = fminf(fmaxf(se11 * ratio, 1.0e-4f), 1.0f - 1.0e-4f);
  T.nf = (float)n_round;
  T.inv_nf = 1.0f / T.nf;
  T.k = k;
  double acc = 0.0;
  float p00 = 1.0f, p01 = 1.0f, p11 = 1.0f;
  for (int e = 0; e <= k; ++e) {
    T.pw00[e] = p00; T.pw01[e] = p01; T.pw11[e] = p11;
    p00 *= s00; p01 *= s01; p11 *= s11;
    T.a[e] = 1.0f / (1.0f + expf(-alpha[e]));
    if (e == 0) T.lf[0] = 0.0f;
    else { acc += log((double)e); T.lf[e] = (float)acc; }
  }
}

// ---------------------------------------------------------------------------
// Kernel A: triplet sums E0..E3 over indices_seq rows.
// Per-block reduction uses V_WMMA_F32_16X16X4_F32 with B = ones:
// D[m,n] = sum_k A[m,k] (row sums), so only the documented A and C/D layouts
// matter. 4 chunks of 64 values are chained through the C accumulator.
// ---------------------------------------------------------------------------
__global__ void __launch_bounds__(TPB)
triplet_kernel(const float* alpha, const float* pse00, const float* pse01,
               const float* pse11, const float* pssum, const int* iseq,
               const int* pnr, const int* pk, int Nt, float* ws) {
  __shared__ Tabs T;
  __shared__ float part[4][TPB];
  const int tid = threadIdx.x;
  if (tid == 0) build_tabs(alpha, *pse00, *pse01, *pse11, *pssum, *pnr, *pk, T);
  __syncthreads();

  float e0 = 0.0f, e1 = 0.0f, e2 = 0.0f, e3 = 0.0f;
  const int idx = blockIdx.x * TPB + tid;
  if (idx < Nt) {
    const int* ei = iseq + (long long)idx * 14;
    __builtin_prefetch(ei, 0, 1);
    const float nf = T.nf;
    const int x0  = ei[0];
    const int x00 = ei[2], x01 = ei[3], x10 = ei[4], x11 = ei[5];
    const int x000 = ei[6], x001 = ei[7], x010 = ei[8], x011 = ei[9];
    const int x100 = ei[10], x101 = ei[11], x110 = ei[12], x111 = ei[13];

    const float p12 = T.pw00[x00] * T.pw01[x01] * T.pw01[x10] * T.pw11[x11];
    const float p13 = T.pw00[x000 + x010] * T.pw01[x001 + x011] *
                      T.pw01[x100 + x110] * T.pw11[x101 + x111];
    const float p23 = T.pw00[x000 + x100] * T.pw01[x001 + x101] *
                      T.pw01[x010 + x110] * T.pw11[x011 + x111];

    const float logn = T.lf[T.k] - (T.lf[x000] + T.lf[x001] + T.lf[x010] + T.lf[x011] +
                                    T.lf[x100] + T.lf[x101] + T.lf[x110] + T.lf[x111]);
    const float ntr = expf(logn) / 6.0f;

    const float a1 = T.a[x0];
    const float a2 = T.a[x00 + x10];
    const float a3 = T.a[x000 + x010 + x100 + x110];

    float ps[3]  = {p12, p13, p23};
    float als[3] = {a3, a2, a1};
    float Qs[3]  = {a1 * a2, a1 * a3, a2 * a3};
    float rr[3], pm[3];
    for (int i = 0; i < 3; ++i) {
      float Rn = 1.0f - powf(1.0f - ps[i], T.inv_nf);
      rr[i] = fminf(Rn / Qs[i], 1.0f);
      pm[i] = fmaxf(1.0f - ((1.0f - ps[i]) / powf(1.0f - Qs[i], nf) + EPSF), 0.0f);
    }
    // stable ascending 3-sort on rr (strict > keeps stability like jnp.argsort)
#define SW3(i, j)                                                           \
    if (rr[i] > rr[j]) {                                                    \
      float t_;                                                             \
      t_ = rr[i]; rr[i] = rr[j]; rr[j] = t_;                                \
      t_ = pm[i]; pm[i] = pm[j]; pm[j] = t_;                                \
      t_ = als[i]; als[i] = als[j]; als[j] = t_;                            \
    }
    SW3(0, 1) SW3(1, 2) SW3(0, 1)
#undef SW3
    const float rmin = rr[0], rmid = rr[1], rmax = rr[2];
    const float pmn = pm[0], pmm = pm[1], pmx = pm[2];
    const float amn = als[0], amd = als[1], amx = als[2];

    const float pg3  = amn * amd * amx;
    const float pg2n = (1.0f - amn) * amd * amx;
    const float pg2m = amn * (1.0f - amd) * amx;
    const float pg2x = amn * amd * (1.0f - amx);
    const float p3  = pg3 * rmin;
    const float p2  = pg3 * (rmid - rmin);
    const float p1n = pg2n * rmin;
    const float p1m = pg2m * rmid;
    const float p1x = pg2x * rmax + pg3 * (rmax - rmid);
    const float p0  = 1.0f - p3 - p2 - p1n - p1m - p1x;
    const float qA = p0 + p2 + p1m + p1x;
    const float qB = p0 + p1n + p1x;
    const float qC = p0 + p1n + p1m;
    const float sA = powf(qA, nf) * (1.0f - pmn);
    const float sB = powf(qB, nf) * (1.0f - pmm);
    const float sC = powf(qC, nf) * (1.0f - pmx);
    const float sAB = powf(p0 + p1x, nf) * (1.0f - pmn) * (1.0f - pmm);
    const float sAC = powf(p0 + p1m, nf) * (1.0f - pmn) * (1.0f - pmx);
    const float sBC = powf(p0 + p1n, nf) * (1.0f - pmm) * (1.0f - pmx);
    const float sABC = powf(p0, nf) * (1.0f - pmn) * (1.0f - pmm) * (1.0f - pmx);
    const float p_all3 = 1.0f - sA - sB - sC + sAB + sAC + sBC - sABC;
    const float p_none = sABC;
    const float p_one = (sBC - sABC) + (sAC - sABC) + (sAB - sABC);
    const float p_two = sA + sB + sC - 2.0f * (sAB + sAC + sBC) + 3.0f * sABC;

    e0 = ntr * p_none; e1 = ntr * p_one; e2 = ntr * p_two; e3 = ntr * p_all3;
  }
  part[0][tid] = e0; part[1][tid] = e1; part[2][tid] = e2; part[3][tid] = e3;
  __syncthreads();

  // Wave-0 WMMA reduction of 256 partials per quantity.
  if (tid < 32) {
    const int lane = tid;
    v2f bones; bones.x = 1.0f; bones.y = 1.0f;           // B = ones(4x16)
    // A layout (32-bit 16x4): lanes 0-15 hold A[m,0..1], lanes 16-31 A[m,2..3]
    const int abase = 4 * (lane & 15) + 2 * (lane >> 4);
    for (int q = 0; q < 4; ++q) {
      v8f c = {0.0f, 0.0f, 0.0f, 0.0f, 0.0f, 0.0f, 0.0f, 0.0f};
      for (int chunk = 0; chunk < 4; ++chunk) {
        v2f av;
        av.x = part[q][chunk * 64 + abase];
        av.y = part[q][chunk * 64 + abase + 1];
        c = __builtin_amdgcn_wmma_f32_16x16x4_f32(
            /*neg_a=*/false, av, /*neg_b=*/false, bones,
            /*c_mod=*/(short)0, c, /*reuse_a=*/false, /*reuse_b=*/false);
      }
      // lane L<16 holds D rows 0..7 in c[0..7]; lanes 16..31 hold rows 8..15
      float t = c[0] + c[1] + c[2] + c[3] + c[4] + c[5] + c[6] + c[7];
      if (lane == 0)  ws[blockIdx.x * 8 + q * 2 + 0] = t;
      if (lane == 16) ws[blockIdx.x * 8 + q * 2 + 1] = t;
    }
  }
}

// ---------------------------------------------------------------------------
// Kernel B: iso term. Rows group by popcount(i) (13 types); each row has
// <= (pi+1)*(k-pi+1) column classes (c1,c0) with multiplicity
// C(pi,c1)*C(k-pi,c0). Sorted-cumprod sum and Prem product computed per class.
// ---------------------------------------------------------------------------
__global__ void iso_kernel(const float* alpha, const float* pse00, const float* pse01,
                           const float* pse11, const float* pssum, const int* pnr,
                           const int* pk, float* ws_iso) {
  __shared__ Tabs T;
  const int tid = threadIdx.x;
  if (tid == 0) build_tabs(alpha, *pse00, *pse01, *pse11, *pssum, *pnr, *pk, T);
  __syncthreads();
  const int k = T.k;
  if (tid > k) return;

  double C[KMAX + 1][KMAX + 1];
  for (int n = 0; n <= k; ++n)
    for (int r = 0; r <= k; ++r) C[n][r] = 0.0;
  for (int n = 0; n <= k; ++n) {
    C[n][0] = 1.0;
    for (int r = 1; r <= n; ++r) C[n][r] = C[n - 1][r - 1] + C[n - 1][r];
  }

  const int pi = tid;
  const float alpha_i = T.a[k - pi];
  const float nf = T.nf;

  float ep[(KMAX + 1) * (KMAX + 1)];
  float sp[(KMAX + 1) * (KMAX + 1)];
  float ml[(KMAX + 1) * (KMAX + 1)];
  int nc = 0;
  float prodPrem = 1.0f;

  for (int c1 = 0; c1 <= pi; ++c1) {
    for (int c0 = 0; c0 <= k - pi; ++c0) {
      const float mult = (float)(C[pi][c1] * C[k - pi][c0]);
      const int n11 = c1, n10 = pi - c1, n01 = c0, n00 = k - pi - c0;
      const float Pf = T.pw00[n00] * T.pw01[n01] * T.pw01[n10] * T.pw11[n11];
      const float Rnum = 1.0f - powf(1.0f - Pf, T.inv_nf);
      const float aj = T.a[k - (c1 + c0)];
      const float Qf = alpha_i * aj;
      const float epv = fminf(Rnum / (Qf + EPSF), 1.0f);
      const float premv = fmaxf(1.0f - ((1.0f - Pf) / powf(1.0f - Qf, nf) + EPSF), 0.0f);
      // diagonal class (j == i) has multiplicity 1: excluded from Prem product
      if (!(c1 == pi && c0 == 0)) prodPrem *= powf(1.0f - premv, mult);
      ep[nc] = epv; sp[nc] = aj; ml[nc] = mult; ++nc;
    }
  }

  // stable insertion sort, descending by ep (equal keys keep enumeration order)
  for (int i = 1; i < nc; ++i) {
    const float ke = ep[i], ks = sp[i], km = ml[i];
    int j = i - 1;
    while (j >= 0 && ep[j] < ke) {
      ep[j + 1] = ep[j]; sp[j + 1] = sp[j]; ml[j + 1] = ml[j];
      --j;
    }
    ep[j + 1] = ke; sp[j + 1] = ks; ml[j + 1] = km;
  }

  // exclusive-cumprod weighted sum, collapsed per class (exact geometric form)
  float cp = 1.0f, pc = 0.0f;
  for (int i = 0; i < nc; ++i) {
    const float qm = powf(1.0f - sp[i], ml[i]);
    pc += cp * ep[i] * (1.0f - qm);
    cp *= qm;
  }
  pc *= alpha_i;
  const float pirow = powf(1.0f - pc, nf) * prodPrem;
  ws_iso[pi] = (float)C[k][pi] * pirow; // C(k,pi) rows share this value
}

// ---------------------------------------------------------------------------
// Kernel C: deterministic fixed-order final reduction into d_out[0..4].
// ---------------------------------------------------------------------------
__global__ void finalize_kernel(const float* ws, int NB, int isoOff, const int* pk,
                                float* out) {
  const int t = threadIdx.x;
  if (t < 4) {
    float s = 0.0f;
    for (int b = 0; b < NB; ++b)
      s += ws[b * 8 + 2 * t] + ws[b * 8 + 2 * t + 1];
    out[t] = s;
  } else if (t == 4) {
    const int k = *pk;
    float s = 0.0f;
    for (int i = 0; i <= k; ++i) s += ws[isoOff + i];
    out[4] = s;
  }
}

extern "C" void kernel_launch(void* const* d_in, const int* in_sizes, int n_in,
                              void* d_out, int out_size, void* d_ws, size_t ws_size,
                              hipStream_t stream) {
  const float* alpha = (const float*)d_in[0];
  const float* s00  = (const float*)d_in[1];
  const float* s01  = (const float*)d_in[2];
  const float* s11  = (const float*)d_in[3];
  const float* ssum = (const float*)d_in[4];
  const int* iseq   = (const int*)d_in[5];
  // d_in[6] (zero_index) not needed: zero_index[i] == k - popcount(i)
  const int* pnr = (const int*)d_in[7];
  const int* pk  = (const int*)d_in[8];

  const int Nt = in_sizes[5] / 14;
  const int NB = (Nt + TPB - 1) / TPB;
  float* ws = (float*)d_ws;
  const int isoOff = NB * 8;

  triplet_kernel<<<NB, TPB, 0, stream>>>(alpha, s00, s01, s11, ssum, iseq, pnr, pk,
                                         Nt, ws);
  iso_kernel<<<1, 32, 0, stream>>>(alpha, s00, s01, s11, ssum, pnr, pk, ws + isoOff);
  finalize_kernel<<<1, 32, 0, stream>>>(ws, NB, isoOff, pk, (float*)d_out);
}